// RegionProposalNetwork_3178275799520
// MI455X (gfx1250) — compile-verified
//
#include <hip/hip_runtime.h>
#include <hip/hip_bf16.h>

typedef __attribute__((ext_vector_type(2))) float v2f;
typedef __attribute__((ext_vector_type(8))) float v8f;
typedef unsigned int u32;
typedef unsigned long long u64;

#define H 50
#define W 76
#define HW 3800      // 50*76
#define MPAD 3808    // 238*16
#define C 512
#define NA 9
#define NSCORE 34200 // 3800*9
#define NSORT 65536
#define PRE_TOPK 10000
#define POST_TOPK 2000
#define NMS_THR_F 0.7f
#define COLW 313     // ceil(10000/32)

// ---------------------------------------------------------------------------
// 0. init: pad sort keys with +inf-equivalent, zero the output buffer
// ---------------------------------------------------------------------------
__global__ void rpn_init(u64* __restrict__ keys, float* __restrict__ out) {
    int i = blockIdx.x * blockDim.x + threadIdx.x;
    if (i < NSORT && i >= NSCORE) keys[i] = ~0ull;
    if (i < PRE_TOPK) out[i] = 0.0f;
}

// ---------------------------------------------------------------------------
// 1. repack conv weights [O][I][3][3] -> [p][I][O] for coalesced GEMM-B loads
// ---------------------------------------------------------------------------
__global__ void repack_w(const float* __restrict__ w, float* __restrict__ wT) {
    int t = blockIdx.x * blockDim.x + threadIdx.x;
    if (t >= 9 * C * C) return;
    int p   = t / (C * C);
    int rem = t - p * (C * C);
    int ci  = rem >> 9;        // /512
    int co  = rem & 511;
    wT[t] = w[co * (C * 9) + ci * 9 + p];
}

// ---------------------------------------------------------------------------
// 2. 3x3 conv as implicit GEMM via V_WMMA_F32_16X16X4_F32 (+bias, ReLU)
//    grid (238, 4), block 256 (8 waves); M-tile 16, N-tile 128 (16/wave)
// ---------------------------------------------------------------------------
__global__ __launch_bounds__(256) void conv3x3_wmma(
        const float* __restrict__ fm,   // [512][50][76]
        const float* __restrict__ wT,   // [9][512][512]
        const float* __restrict__ bias, // [512]
        float* __restrict__ feat)       // [3808][512]
{
    __shared__ float At[16][36];   // im2col tile, padded row (36%64 banks -> conflict-free)
    __shared__ float Bt[32][128];

    const int tid  = threadIdx.x;
    const int wv   = tid >> 5;
    const int lane = tid & 31;
    const int half = lane >> 4;    // 0: K pair {0,1}, 1: K pair {2,3}
    const int l16  = lane & 15;
    const int m0   = blockIdx.x * 16;
    const int n0   = blockIdx.y * 128;
    const int nw   = wv * 16;

    v8f acc = {};

    for (int p = 0; p < 9; ++p) {
        const int dy = p / 3 - 1;
        const int dx = p % 3 - 1;
        const float* wp = wT + p * (C * C) + n0;

        for (int c0 = 0; c0 < C; c0 += 32) {
            __syncthreads();
            // stage A: 16 rows (M) x 32 channels, zero-padded im2col
            for (int e = tid; e < 512; e += 256) {
                int r = e >> 5, cc = e & 31;
                int m = m0 + r;
                int y = m / W, x = m - y * W;
                int iy = y + dy, ix = x + dx;
                float v = 0.0f;
                if (m < HW && (unsigned)iy < (unsigned)H && (unsigned)ix < (unsigned)W)
                    v = fm[(c0 + cc) * HW + iy * W + ix];
                At[r][cc] = v;
            }
            // stage B: 32 K x 128 N (fully coalesced from repacked weights)
            for (int e = tid; e < 4096; e += 256) {
                int kk = e >> 7, n = e & 127;
                Bt[kk][n] = wp[(c0 + kk) * C + n];
            }
            __syncthreads();

            #pragma unroll
            for (int kk = 0; kk < 32; kk += 4) {
                v2f a, b;
                // A 16x4 f32 layout: lanes 0-15 hold K={kk,kk+1}, lanes 16-31 K={kk+2,kk+3}
                a.x = At[l16][kk + 2 * half];
                a.y = At[l16][kk + 2 * half + 1];
                // B 4x16 f32 layout: mirrored halves over K, lane%16 = N
                b.x = Bt[kk + 2 * half][nw + l16];
                b.y = Bt[kk + 2 * half + 1][nw + l16];
                acc = __builtin_amdgcn_wmma_f32_16x16x4_f32(
                        false, a, false, b, (short)0, acc, false, false);
            }
        }
    }

    // C/D layout: VGPR r -> M = r (+8 for upper half-lanes), lane%16 -> N
    const int n  = n0 + nw + l16;
    const float bn = bias[n];
    #pragma unroll
    for (int r = 0; r < 8; ++r) {
        int m = m0 + r + half * 8;
        float v = acc[r] + bn;
        feat[m * C + n] = v > 0.0f ? v : 0.0f;
    }
}

// ---------------------------------------------------------------------------
// 3. heads: 1x1 convs + sigmoid + anchors + regression + clip + sort keys
//    one block per spatial position m
// ---------------------------------------------------------------------------
__global__ __launch_bounds__(128) void rpn_heads(
        const float* __restrict__ feat,
        const float* __restrict__ cls_w, const float* __restrict__ cls_b,
        const float* __restrict__ bbox_w, const float* __restrict__ bbox_b,
        float* __restrict__ scores, float* __restrict__ proposals,
        u64* __restrict__ keys)
{
    __shared__ float f[C];
    __shared__ float outv[45];
    const int m   = blockIdx.x;
    const int tid = threadIdx.x;

    for (int c = tid; c < C; c += 128) f[c] = feat[m * C + c];
    __syncthreads();

    if (tid < 45) {
        const float* wrow = (tid < NA) ? (cls_w + tid * C) : (bbox_w + (tid - NA) * C);
        float s = 0.0f;
        for (int c = 0; c < C; ++c) s += f[c] * wrow[c];
        s += (tid < NA) ? cls_b[tid] : bbox_b[tid - NA];
        outv[tid] = s;
    }
    __syncthreads();

    if (tid < NA) {
        const int a = tid;
        const float score = 1.0f / (1.0f + __expf(-outv[a]) * 0.0f + expf(-outv[a]) * 1.0f - 1.0f + 1.0f) ; // placeholder avoided below
        (void)score;
        float sc = 1.0f / (1.0f + expf(-outv[a]));
        float t0 = outv[NA + a * 4 + 0];
        float t1 = outv[NA + a * 4 + 1];
        float t2 = outv[NA + a * 4 + 2];
        float t3 = outv[NA + a * 4 + 3];

        // anchors: stride 16; a = ratio_idx*3 + scale_idx
        const float ratios[3] = {0.5f, 1.0f, 2.0f};
        const float scalesv[3] = {128.0f, 256.0f, 512.0f};
        int ri = a / 3, si = a % 3;
        int y = m / W, x = m - y * W;
        float sx = (float)x * 16.0f, sy = (float)y * 16.0f;
        float hr = sqrtf(ratios[ri]);
        float wr = 1.0f / hr;
        float bw = rintf(wr * scalesv[si] * 0.5f);   // RNE == jnp.round
        float bh = rintf(hr * scalesv[si] * 0.5f);
        float ax1 = sx - bw, ay1 = sy - bh, ax2 = sx + bw, ay2 = sy + bh;

        float aw = ax2 - ax1, ah = ay2 - ay1;
        float cx = ax1 + 0.5f * aw, cy = ay1 + 0.5f * ah;
        float pcx = t0 * aw + cx, pcy = t1 * ah + cy;
        float pw = expf(t2) * aw, ph = expf(t3) * ah;
        float x1 = pcx - 0.5f * pw, y1 = pcy - 0.5f * ph;
        float x2 = pcx + 0.5f * pw, y2 = pcy + 0.5f * ph;
        x1 = fminf(fmaxf(x1, 0.0f), 1216.0f);
        y1 = fminf(fmaxf(y1, 0.0f), 800.0f);
        x2 = fminf(fmaxf(x2, 0.0f), 1216.0f);
        y2 = fminf(fmaxf(y2, 0.0f), 800.0f);

        int i = m * NA + a;
        proposals[i * 4 + 0] = x1;
        proposals[i * 4 + 1] = y1;
        proposals[i * 4 + 2] = x2;
        proposals[i * 4 + 3] = y2;
        scores[i] = sc;

        // descending-score key, ascending-index ties (matches lax.top_k)
        u32 sb = __float_as_uint(sc);
        sb = (sb & 0x80000000u) ? ~sb : (sb | 0x80000000u); // ascending transform
        u32 desc = ~sb;
        keys[i] = ((u64)desc << 32) | (u32)i;
    }
}

// ---------------------------------------------------------------------------
// 4. bitonic sort step over 65536 64-bit keys (ascending)
// ---------------------------------------------------------------------------
__global__ void bitonic_step(u64* __restrict__ keys, int j, int k) {
    int tid = blockIdx.x * blockDim.x + threadIdx.x;
    int ixj = tid ^ j;
    if (ixj > tid) {
        u64 a = keys[tid];
        u64 b = keys[ixj];
        bool up = ((tid & k) == 0);
        if (up ? (a > b) : (a < b)) {
            keys[tid] = b;
            keys[ixj] = a;
        }
    }
}

// ---------------------------------------------------------------------------
// 5. gather sorted top-10000 boxes and scores
// ---------------------------------------------------------------------------
__global__ void gather_topk(const u64* __restrict__ keys,
                            const float* __restrict__ scores,
                            const float* __restrict__ proposals,
                            float* __restrict__ props, float* __restrict__ tscores) {
    int r = blockIdx.x * blockDim.x + threadIdx.x;
    if (r >= PRE_TOPK) return;
    u32 idx = (u32)(keys[r] & 0xffffffffu);
    tscores[r] = scores[idx];
    props[r * 4 + 0] = proposals[idx * 4 + 0];
    props[r * 4 + 1] = proposals[idx * 4 + 1];
    props[r * 4 + 2] = proposals[idx * 4 + 2];
    props[r * 4 + 3] = proposals[idx * 4 + 3];
}

// ---------------------------------------------------------------------------
// 6. NMS IoU bitmask: grid (313, 313), 32 threads (one wave32)
// ---------------------------------------------------------------------------
__global__ __launch_bounds__(32) void nms_mask(const float* __restrict__ props,
                                               u32* __restrict__ mask) {
    __shared__ float cb[32][4];
    const int bi = blockIdx.x, bj = blockIdx.y;
    const int t = threadIdx.x;
    const int i = bi * 32 + t;
    const int j0 = bj * 32;
    const int jt = j0 + t;

    cb[t][0] = (jt < PRE_TOPK) ? props[jt * 4 + 0] : 0.0f;
    cb[t][1] = (jt < PRE_TOPK) ? props[jt * 4 + 1] : 0.0f;
    cb[t][2] = (jt < PRE_TOPK) ? props[jt * 4 + 2] : 0.0f;
    cb[t][3] = (jt < PRE_TOPK) ? props[jt * 4 + 3] : 0.0f;
    __syncthreads();

    if (i >= PRE_TOPK) return;
    float x1 = props[i * 4 + 0], y1 = props[i * 4 + 1];
    float x2 = props[i * 4 + 2], y2 = props[i * 4 + 3];
    float area_i = (x2 - x1) * (y2 - y1);

    u32 bits = 0;
    #pragma unroll 4
    for (int jj = 0; jj < 32; ++jj) {
        int j = j0 + jj;
        if (j < PRE_TOPK && j != i) {
            float ltx = fmaxf(x1, cb[jj][0]);
            float lty = fmaxf(y1, cb[jj][1]);
            float rbx = fminf(x2, cb[jj][2]);
            float rby = fminf(y2, cb[jj][3]);
            float iw = fmaxf(rbx - ltx, 0.0f);
            float ih = fmaxf(rby - lty, 0.0f);
            float inter = iw * ih;
            float area_j = (cb[jj][2] - cb[jj][0]) * (cb[jj][3] - cb[jj][1]);
            float iou = inter / (area_i + area_j - inter); // NaN -> not suppressed (matches jnp)
            if (iou > NMS_THR_F) bits |= (1u << jj);
        }
    }
    mask[(size_t)i * COLW + bj] = bits;
}

// ---------------------------------------------------------------------------
// 7. greedy NMS reduction: single wave32, remv bitset in registers (10 words/lane)
// ---------------------------------------------------------------------------
__global__ __launch_bounds__(32) void nms_reduce(const u32* __restrict__ mask,
                                                 u32* __restrict__ keep) {
    const int lane = threadIdx.x;
    u32 remv[10];
    #pragma unroll
    for (int q = 0; q < 10; ++q) remv[q] = 0;

    #pragma unroll
    for (int s = 0; s < 10; ++s) {
        for (int ii = 0; ii < 1024; ++ii) {
            int i = s * 1024 + ii;
            if (i >= PRE_TOPK) break;
            int w = i >> 5;                              // word index; lane w&31 owns slot s
            u32 val = (u32)__shfl((int)remv[s], w & 31);
            bool kp = ((val >> (i & 31)) & 1u) == 0u;
            if (lane == 0) keep[i] = kp ? 1u : 0u;
            if (kp) {
                const u32* row = mask + (size_t)i * COLW;
                #pragma unroll
                for (int q = 0; q < 10; ++q) {
                    int wi = q * 32 + lane;
                    if (wi < COLW) remv[q] |= row[wi];
                }
            }
        }
    }
}

// ---------------------------------------------------------------------------
// 8. compaction: ballot+popcount ranks, first 2000 kept in ascending index order
//    out layout: [2000*4 boxes][2000 scores] (tuple flattened in return order)
// ---------------------------------------------------------------------------
__global__ __launch_bounds__(32) void nms_finalize(const u32* __restrict__ keep,
                                                   const float* __restrict__ props,
                                                   const float* __restrict__ tscores,
                                                   float* __restrict__ out) {
    const int lane = threadIdx.x;
    u32 total = 0;
    for (int base = 0; base < PRE_TOPK; base += 32) {
        int i = base + lane;
        bool k = keep[i] != 0u;
        u32 bal = (u32)__ballot(k);
        u32 rank = total + __popc(bal & ((1u << lane) - 1u));
        if (k && rank < POST_TOPK) {
            out[rank * 4 + 0] = props[i * 4 + 0];
            out[rank * 4 + 1] = props[i * 4 + 1];
            out[rank * 4 + 2] = props[i * 4 + 2];
            out[rank * 4 + 3] = props[i * 4 + 3];
            out[POST_TOPK * 4 + rank] = tscores[i];
        }
        total += __popc(bal);
    }
}

// ---------------------------------------------------------------------------
extern "C" void kernel_launch(void* const* d_in, const int* in_sizes, int n_in,
                              void* d_out, int out_size, void* d_ws, size_t ws_size,
                              hipStream_t stream) {
    // inputs per setup_inputs() order
    const float* fm     = (const float*)d_in[1];  // feature_map [512][50][76]
    const float* conv_w = (const float*)d_in[2];  // [512][512][3][3]
    const float* conv_b = (const float*)d_in[3];
    const float* cls_w  = (const float*)d_in[4];  // [9][512]
    const float* cls_b  = (const float*)d_in[5];
    const float* bbox_w = (const float*)d_in[6];  // [36][512]
    const float* bbox_b = (const float*)d_in[7];
    float* out = (float*)d_out;                   // 2000*4 boxes + 2000 scores

    // workspace carve-up
    char* wsp = (char*)d_ws;
    auto carve = [&](size_t bytes) -> char* {
        char* p = wsp;
        wsp += (bytes + 255) & ~(size_t)255;
        return p;
    };
    float* wT        = (float*)carve((size_t)9 * C * C * 4);
    float* feat      = (float*)carve((size_t)MPAD * C * 4);
    float* scores    = (float*)carve((size_t)NSCORE * 4);
    float* proposals = (float*)carve((size_t)NSCORE * 16);
    u64*   keys      = (u64*)  carve((size_t)NSORT * 8);
    float* propsS    = (float*)carve((size_t)PRE_TOPK * 16);
    float* tscores   = (float*)carve((size_t)PRE_TOPK * 4);
    u32*   mask      = (u32*)  carve((size_t)PRE_TOPK * COLW * 4);
    u32*   keep      = (u32*)  carve((size_t)PRE_TOPK * 4);

    // 0. init keys tail + zero output
    rpn_init<<<NSORT / 256, 256, 0, stream>>>(keys, out);
    // 1. repack weights
    repack_w<<<(9 * C * C + 255) / 256, 256, 0, stream>>>(conv_w, wT);
    // 2. conv3x3 + ReLU (WMMA f32)
    conv3x3_wmma<<<dim3(MPAD / 16, 4), 256, 0, stream>>>(fm, wT, conv_b, feat);
    // 3. heads + proposals + sort keys
    rpn_heads<<<HW, 128, 0, stream>>>(feat, cls_w, cls_b, bbox_w, bbox_b,
                                      scores, proposals, keys);
    // 4. bitonic sort (65536 keys, ascending = descending score)
    for (int k = 2; k <= NSORT; k <<= 1)
        for (int j = k >> 1; j >= 1; j >>= 1)
            bitonic_step<<<NSORT / 256, 256, 0, stream>>>(keys, j, k);
    // 5. gather top-10000
    gather_topk<<<(PRE_TOPK + 255) / 256, 256, 0, stream>>>(keys, scores, proposals,
                                                            propsS, tscores);
    // 6. IoU mask
    nms_mask<<<dim3(COLW, COLW), 32, 0, stream>>>(propsS, mask);
    // 7. greedy serial reduction (one wave)
    nms_reduce<<<1, 32, 0, stream>>>(mask, keep);
    // 8. compact kept boxes/scores into output
    nms_finalize<<<1, 32, 0, stream>>>(keep, propsS, tscores, out);
}